// Generate_Correspondence_16406775071172
// MI455X (gfx1250) — compile-verified
//
#include <hip/hip_runtime.h>
#include <hip/hip_bf16.h>
#include <math.h>

typedef __attribute__((ext_vector_type(2))) float v2f;
typedef __attribute__((ext_vector_type(8))) float v8f;

#define K_FX     384.996f
#define K_FY     384.996f
#define K_INV_FX (1.0f / 384.996f)
#define K_INV_FY (1.0f / 384.996f)
#define K_CX     325.85f
#define K_CY     237.646f
#define K_MARGIN 0.003f
#define IMG_W 640
#define IMG_H 480

// ---------------------------------------------------------------------------
// Setup kernel: T = inv(pose_B) * pose_A  (general 4x4 inverse, like jnp.linalg.inv)
// ---------------------------------------------------------------------------
__device__ static void inv4x4(const float* m, float* inv) {
    inv[0]  =  m[5]*m[10]*m[15] - m[5]*m[11]*m[14] - m[9]*m[6]*m[15] + m[9]*m[7]*m[14] + m[13]*m[6]*m[11] - m[13]*m[7]*m[10];
    inv[4]  = -m[4]*m[10]*m[15] + m[4]*m[11]*m[14] + m[8]*m[6]*m[15] - m[8]*m[7]*m[14] - m[12]*m[6]*m[11] + m[12]*m[7]*m[10];
    inv[8]  =  m[4]*m[9]*m[15]  - m[4]*m[11]*m[13] - m[8]*m[5]*m[15] + m[8]*m[7]*m[13] + m[12]*m[5]*m[11] - m[12]*m[7]*m[9];
    inv[12] = -m[4]*m[9]*m[14]  + m[4]*m[10]*m[13] + m[8]*m[5]*m[14] - m[8]*m[6]*m[13] - m[12]*m[5]*m[10] + m[12]*m[6]*m[9];
    inv[1]  = -m[1]*m[10]*m[15] + m[1]*m[11]*m[14] + m[9]*m[2]*m[15] - m[9]*m[3]*m[14] - m[13]*m[2]*m[11] + m[13]*m[3]*m[10];
    inv[5]  =  m[0]*m[10]*m[15] - m[0]*m[11]*m[14] - m[8]*m[2]*m[15] + m[8]*m[3]*m[14] + m[12]*m[2]*m[11] - m[12]*m[3]*m[10];
    inv[9]  = -m[0]*m[9]*m[15]  + m[0]*m[11]*m[13] + m[8]*m[1]*m[15] - m[8]*m[3]*m[13] - m[12]*m[1]*m[11] + m[12]*m[3]*m[9];
    inv[13] =  m[0]*m[9]*m[14]  - m[0]*m[10]*m[13] - m[8]*m[1]*m[14] + m[8]*m[2]*m[13] + m[12]*m[1]*m[10] - m[12]*m[2]*m[9];
    inv[2]  =  m[1]*m[6]*m[15]  - m[1]*m[7]*m[14]  - m[5]*m[2]*m[15] + m[5]*m[3]*m[14] + m[13]*m[2]*m[7]  - m[13]*m[3]*m[6];
    inv[6]  = -m[0]*m[6]*m[15]  + m[0]*m[7]*m[14]  + m[4]*m[2]*m[15] - m[4]*m[3]*m[14] - m[12]*m[2]*m[7]  + m[12]*m[3]*m[6];
    inv[10] =  m[0]*m[5]*m[15]  - m[0]*m[7]*m[13]  - m[4]*m[1]*m[15] + m[4]*m[3]*m[13] + m[12]*m[1]*m[7]  - m[12]*m[3]*m[5];
    inv[14] = -m[0]*m[5]*m[14]  + m[0]*m[6]*m[13]  + m[4]*m[1]*m[14] - m[4]*m[2]*m[13] - m[12]*m[1]*m[6]  + m[12]*m[2]*m[5];
    inv[3]  = -m[1]*m[6]*m[11]  + m[1]*m[7]*m[10]  + m[5]*m[2]*m[11] - m[5]*m[3]*m[10] - m[9]*m[2]*m[7]   + m[9]*m[3]*m[6];
    inv[7]  =  m[0]*m[6]*m[11]  - m[0]*m[7]*m[10]  - m[4]*m[2]*m[11] + m[4]*m[3]*m[10] + m[8]*m[2]*m[7]   - m[8]*m[3]*m[6];
    inv[11] = -m[0]*m[5]*m[11]  + m[0]*m[7]*m[9]   + m[4]*m[1]*m[11] - m[4]*m[3]*m[9]  - m[8]*m[1]*m[7]   + m[8]*m[3]*m[5];
    inv[15] =  m[0]*m[5]*m[10]  - m[0]*m[6]*m[9]   - m[4]*m[1]*m[10] + m[4]*m[2]*m[9]  + m[8]*m[1]*m[6]   - m[8]*m[2]*m[5];
    float det = m[0]*inv[0] + m[1]*inv[4] + m[2]*inv[8] + m[3]*inv[12];
    float rdet = 1.0f / det;
    for (int i = 0; i < 16; ++i) inv[i] *= rdet;
}

__global__ void setup_T_kernel(const float* __restrict__ poseA,
                               const float* __restrict__ poseB,
                               float* __restrict__ Tws) {
    if (threadIdx.x == 0 && blockIdx.x == 0) {
        float invB[16];
        inv4x4(poseB, invB);
        // T = invB * poseA
        for (int i = 0; i < 4; ++i)
            for (int j = 0; j < 4; ++j) {
                float s = 0.f;
                for (int k = 0; k < 4; ++k) s += invB[i*4 + k] * poseA[k*4 + j];
                Tws[i*4 + j] = s;
            }
    }
}

// ---------------------------------------------------------------------------
// Main kernel: one wave handles 16 points per iteration via one
// v_wmma_f32_16x16x4_f32 (D = T(16x4 padded) x Points(4x16)).
//
// f32 A layout (16x4): lanes 0-15 hold row M=lane with VGPR0=K0, VGPR1=K1;
//                      lanes 16-31 hold VGPR0=K2, VGPR1=K3.
// f32 B layout (4x16): lanes 0-15 supply rows K0/K1 (x,y of point n=lane);
//                      lanes 16-31 supply rows K2/K3 (z,1 of point n=lane-16).
//   -> lane j and lane j+16 load the SAME point j (same cache lines, no
//      extra DRAM traffic, no cross-lane shuffles needed to build B).
// f32 D layout (16x16): lane n, VGPR 0/1/2 = rows 0/1/2 of column n
//   -> x_B, y_B, z_B of point n land in lane n's c[0..2]. Epilogue is
//      per-lane SIMD; stores predicated (half==0) AFTER the full-EXEC WMMA.
//
// Perspective divide uses v_rcp_f32 + FMA instead of IEEE-exact division
// (one rcp per point instead of two ~10-op div sequences).
// ---------------------------------------------------------------------------
__global__ void __launch_bounds__(256)
corr_kernel(const float* __restrict__ depthA,
            const float* __restrict__ depthB,
            const int*   __restrict__ uA,
            const int*   __restrict__ vA,
            const float* __restrict__ Tws,
            float*       __restrict__ out,
            int N) {
    const int lane     = threadIdx.x & 31;
    const int j        = lane & 15;       // point slot within the 16-point tile
    const int half     = lane >> 4;       // 0: supplies K0/K1, 1: supplies K2/K3
    const int waveId   = (blockIdx.x * blockDim.x + threadIdx.x) >> 5;
    const int numWaves = (gridDim.x * blockDim.x) >> 5;
    const int stride   = numWaves * 16;

    // A-matrix (constant T, rows 4..15 zero) — loaded once.
    float a0 = 0.f, a1 = 0.f;
    if (j < 4) {
        a0 = Tws[j * 4 + (half ? 2 : 0)];
        a1 = Tws[j * 4 + (half ? 3 : 1)];
    }
    v2f a = {a0, a1};

    for (int base = waveId * 16; base < N; base += stride) {
        int p  = base + j;
        int pc = p < N ? p : (N - 1);           // clamp keeps EXEC full

        // prefetch next tile of the index streams (global_prefetch_b8);
        // clamped address, no branch (speculative prefetch is droppable)
        int pn = pc + stride;
        pn = pn < N ? pn : (N - 1);
        __builtin_prefetch(&uA[pn], 0, 0);
        __builtin_prefetch(&vA[pn], 0, 0);

        int u = uA[pc];
        int v = vA[pc];
        int pix = v * IMG_W + u;
        float zA = depthA[pix];                 // DEPTH_SCALE == 1
        float dB = depthB[pix];

        float uf = (float)u;
        float vf = (float)v;
        float xA = (uf - K_CX) * zA * K_INV_FX;
        float yA = (vf - K_CY) * zA * K_INV_FY;

        v2f b;
        b.x = half ? zA   : xA;                 // K2 : K0
        b.y = half ? 1.0f : yA;                 // K3 : K1

        v8f c = {};
        c = __builtin_amdgcn_wmma_f32_16x16x4_f32(
                /*neg_a=*/false, a, /*neg_b=*/false, b,
                /*c_mod=*/(short)0, c, /*reuse_a=*/false, /*reuse_b=*/false);

        // lanes 0-15: c[0..2] = (x,y,z) of point p in camera-B frame
        float xB = c[0], yB = c[1], zB = c[2];
        float invz = __builtin_amdgcn_rcpf(zB);         // v_rcp_f32
        float uB = fmaf(K_FX * xB, invz, K_CX);
        float vB = fmaf(K_FY * yB, invz, K_CY);
        float uBi = truncf(uB);
        float vBi = truncf(vB);
        bool valid = (zA > 0.f) &
                     (uBi > 0.f) & (uBi < (float)IMG_W) &
                     (vBi > 0.f) & (vBi < (float)IMG_H) &
                     (dB > 0.f) & (dB >= zB - K_MARGIN);

        if (half == 0 && p < N) {
            int o = p * 3;
            out[o + 0] = uB;
            out[o + 1] = vB;
            out[o + 2] = valid ? 1.0f : 0.0f;
        }
    }
}

// ---------------------------------------------------------------------------
// Launch
// inputs: 0 in_A(f32, unused) 1 depth_A(f32) 2 pose_A(f32) 3 in_B(unused)
//         4 depth_B(f32) 5 pose_B(f32) 6 u_A(i32) 7 v_A(i32)
// ---------------------------------------------------------------------------
extern "C" void kernel_launch(void* const* d_in, const int* in_sizes, int n_in,
                              void* d_out, int out_size, void* d_ws, size_t ws_size,
                              hipStream_t stream) {
    const float* depthA = (const float*)d_in[1];
    const float* poseA  = (const float*)d_in[2];
    const float* depthB = (const float*)d_in[4];
    const float* poseB  = (const float*)d_in[5];
    const int*   uA     = (const int*)d_in[6];
    const int*   vA     = (const int*)d_in[7];
    float*       out    = (float*)d_out;
    float*       Tws    = (float*)d_ws;
    const int N = in_sizes[6];

    setup_T_kernel<<<1, 1, 0, stream>>>(poseA, poseB, Tws);

    const int threads = 256;                  // 8 waves/block
    const int blocks  = 2048;                 // 16384 waves, grid-stride
    corr_kernel<<<blocks, threads, 0, stream>>>(depthA, depthB, uA, vA, Tws, out, N);
}